// Classifiergru_70557722739404
// MI455X (gfx1250) — compile-verified
//
#include <hip/hip_runtime.h>
#include <math.h>

#define VOCAB 100
#define EMBED 10
#define MID   100
#define SEQ   550
#define GATES 300          // 3*MID
#define NTILE_N 19         // ceil(300/16)
#define NTILE_M 35         // ceil(550/16)

typedef __attribute__((ext_vector_type(2))) float v2f;
typedef __attribute__((ext_vector_type(8))) float v8f;

__device__ __forceinline__ float sigmoidf_(float x) {
    return 1.0f / (1.0f + __expf(-x));
}

// ---------------------------------------------------------------------------
// Kernel 1: gi_all[t][g] = embed_table[x[t]] . w_ih[g] + b_ih[g]
// One wave per 16x16 tile of [550 x 300]; K=10 padded to 3x4 chunks, three
// chained v_wmma_f32_16x16x4_f32. All loads unconditional (indices clamped:
// out-of-range rows/cols only affect D entries we never store); only the
// K-padding (cols 10,11) is zeroed, via register cndmask. Full tiles take a
// branch-free store path; only the last M tile runs the guarded path.
// ---------------------------------------------------------------------------
__global__ __launch_bounds__(32) void gi_gemm_wmma(
    const int*   __restrict__ x,
    const float* __restrict__ embed_table,
    const float* __restrict__ w_ih,
    const float* __restrict__ b_ih,
    float*       __restrict__ gi_all)
{
    int tile = blockIdx.x;
    int mt = tile / NTILE_N;
    int nt = tile % NTILE_N;
    int lane = threadIdx.x & 31;
    int half = lane >> 4;
    int l16  = lane & 15;
    int cofs = 2 * half;

    int m0 = mt * 16;
    int n0 = nt * 16;

    // A: 16x4 tiles of embedded sequence rows (clamped timestep)
    int trow = m0 + l16;
    int trc  = (trow < SEQ) ? trow : (SEQ - 1);
    const float* eptr = embed_table + x[trc] * EMBED;

    // B: 4x16 tiles of w_ih^T (clamped gate row)
    int ncol = n0 + l16;
    int ncc  = (ncol < GATES) ? ncol : (GATES - 1);
    const float* wptr = w_ih + ncc * EMBED;

    // bias for this output column (clamped index is always safe to load)
    float bias = b_ih[ncc];

    // unconditional b64 loads (all 8-byte aligned: row strides are even floats)
    v2f a0 = *(const v2f*)(eptr + cofs);
    v2f a1 = *(const v2f*)(eptr + 4 + cofs);
    v2f a2 = *(const v2f*)(eptr + 8);        // cols 8,9 (always in-bounds)
    v2f b0 = *(const v2f*)(wptr + cofs);
    v2f b1 = *(const v2f*)(wptr + 4 + cofs);
    v2f b2 = *(const v2f*)(wptr + 8);
    // K-padding: half==1 lanes represent cols 10,11 -> zero
    a2.x = half ? 0.0f : a2.x;  a2.y = half ? 0.0f : a2.y;
    b2.x = half ? 0.0f : b2.x;  b2.y = half ? 0.0f : b2.y;

    v8f acc = {};
    acc = __builtin_amdgcn_wmma_f32_16x16x4_f32(false, a0, false, b0, (short)0, acc, false, false);
    acc = __builtin_amdgcn_wmma_f32_16x16x4_f32(false, a1, false, b1, (short)0, acc, false, false);
    acc = __builtin_amdgcn_wmma_f32_16x16x4_f32(false, a2, false, b2, (short)0, acc, false, false);

    // D: lanes 0-15 -> N=lane, M=vgpr; lanes 16-31 -> N=lane-16, M=vgpr+8
    int n = n0 + l16;
    int mbase = m0 + half * 8;
    if (n < GATES) {
        float* gout = gi_all + (size_t)mbase * GATES + n;
        if (mbase + 8 <= SEQ) {            // full tile: branch-free stores
            #pragma unroll
            for (int m = 0; m < 8; ++m)
                gout[m * GATES] = acc[m] + bias;
        } else {                           // ragged last M tile only
            #pragma unroll
            for (int m = 0; m < 8; ++m)
                if (mbase + m < SEQ) gout[m * GATES] = acc[m] + bias;
        }
    }
}

// ---------------------------------------------------------------------------
// Kernel 2: 550-step GRU recurrence, one resident workgroup (20 waves).
// Waves 0..18 hold their 16x100 slice of w_hh pre-swizzled in VGPRs; per step
// they run a 25-deep chained f32 WMMA. h is replicated across all 16 B
// columns (every column of D then equals gh -> no lane masking anywhere).
// b_hh is folded into the accumulator init. gi is double-buffered in
// registers so its global loads are overlapped with the WMMA phase.
// ---------------------------------------------------------------------------
__global__ __launch_bounds__(640) void gru_seq_wmma(
    const float* __restrict__ hidden0,
    const float* __restrict__ w_hh,
    const float* __restrict__ b_hh,
    const float* __restrict__ gi_all,
    const float* __restrict__ fc_w,
    const float* __restrict__ fc_b,
    float*       __restrict__ out)
{
    __shared__ __align__(16) float h_s[MID];
    __shared__ __align__(16) float gh_s[NTILE_N * 16];  // 304; 300..303 scratch

    int tid  = threadIdx.x;
    int wid  = tid >> 5;
    int lane = tid & 31;
    int half = lane >> 4;
    int l16  = lane & 15;
    int cofs = 2 * half;
    bool mmw = (wid < NTILE_N);

    if (tid < MID) h_s[tid] = hidden0[tid];

    // Preload A operands: 16 rows x 100 cols of w_hh -> 50 VGPRs per lane.
    // Row indices clamped: garbage only lands in gh rows 300..303 (never read).
    v2f   a[25];
    float bias[8];
    if (mmw) {
        int row = wid * 16 + l16;
        const float* wrow = w_hh + ((row < GATES) ? row : 0) * MID;
        #pragma unroll
        for (int k = 0; k < 25; ++k)
            a[k] = *(const v2f*)(wrow + 4 * k + cofs);
        #pragma unroll
        for (int m = 0; m < 8; ++m) {
            int r = wid * 16 + half * 8 + m;
            bias[m] = b_hh[(r < GATES) ? r : 0];   // lane-uniform per half
        }
    }

    // gi double buffer: registers for step t, loads for t+1 issued in step t.
    float g0 = 0.0f, g1 = 0.0f, g2 = 0.0f;
    if (tid < MID) {
        g0 = gi_all[tid];
        g1 = gi_all[MID + tid];
        g2 = gi_all[2 * MID + tid];
    }
    __syncthreads();

    for (int t = 0; t < SEQ; ++t) {
        if (mmw) {
            // gather all B operands first (independent ds_load_b64s), then
            // run the uninterrupted 25-deep WMMA accumulation chain
            v2f bb[25];
            #pragma unroll
            for (int k = 0; k < 25; ++k)
                bb[k] = *(const v2f*)&h_s[4 * k + cofs];  // broadcast load
            v8f acc;
            #pragma unroll
            for (int m = 0; m < 8; ++m) acc[m] = bias[m];
            #pragma unroll
            for (int k = 0; k < 25; ++k)
                acc = __builtin_amdgcn_wmma_f32_16x16x4_f32(
                    false, a[k], false, bb[k], (short)0, acc, false, false);
            // all D columns are identical; lanes 0/16 write rows (two b128s)
            if (l16 == 0) {
                float4* p = (float4*)&gh_s[wid * 16 + half * 8];
                p[0] = make_float4(acc[0], acc[1], acc[2], acc[3]);
                p[1] = make_float4(acc[4], acc[5], acc[6], acc[7]);
            }
        }
        __syncthreads();

        if (tid < MID) {
            float r = sigmoidf_(g0 + gh_s[tid]);
            float z = sigmoidf_(g1 + gh_s[MID + tid]);
            float n = tanhf    (g2 + r * gh_s[2 * MID + tid]);
            h_s[tid] = (1.0f - z) * n + z * h_s[tid];
            if (t + 1 < SEQ) {            // refill for next step; consumed
                const float* gn = gi_all + (t + 1) * GATES;   // after a full
                g0 = gn[tid];                                  // WMMA phase
                g1 = gn[MID + tid];
                g2 = gn[2 * MID + tid];
            }
        }
        __syncthreads();
    }

    if (tid == 0) {
        float s = 0.0f;
        for (int j = 0; j < MID; ++j) {
            float hv = h_s[j];
            s += (hv > 0.0f ? hv : 0.0f) * fc_w[j];
        }
        out[0] = sigmoidf_(s + fc_b[0]);
    }
}

// ---------------------------------------------------------------------------
extern "C" void kernel_launch(void* const* d_in, const int* in_sizes, int n_in,
                              void* d_out, int out_size, void* d_ws, size_t ws_size,
                              hipStream_t stream) {
    const int*   x           = (const int*)  d_in[0];
    const float* hidden      = (const float*)d_in[1];
    const float* embed_table = (const float*)d_in[2];
    const float* w_ih        = (const float*)d_in[3];
    const float* w_hh        = (const float*)d_in[4];
    const float* b_ih        = (const float*)d_in[5];
    const float* b_hh        = (const float*)d_in[6];
    const float* fc_w        = (const float*)d_in[7];
    const float* fc_b        = (const float*)d_in[8];
    float* out    = (float*)d_out;
    float* gi_all = (float*)d_ws;    // needs SEQ*GATES*4 = 660000 B of scratch

    gi_gemm_wmma<<<NTILE_M * NTILE_N, 32, 0, stream>>>(x, embed_table, w_ih, b_ih, gi_all);
    gru_seq_wmma<<<1, 640, 0, stream>>>(hidden, w_hh, b_hh, gi_all, fc_w, fc_b, out);
}